// HierarchicalGCN_33904471835037
// MI455X (gfx1250) — compile-verified
//
#include <hip/hip_runtime.h>
#include <hip/hip_bf16.h>
#include <stdint.h>

#define N_NODES 65536
#define N_EDGES 1048576
#define N_GRAPH 128
#define HF 64
#define NCLS 10
#define NLAY 4
#define EPSV 1e-5f

typedef __attribute__((ext_vector_type(2))) float v2f;
typedef __attribute__((ext_vector_type(8))) float v8f;

// ---------------- generic fill ----------------
__global__ __launch_bounds__(256) void k_fill(float* __restrict__ p, float v, int n) {
  int i = blockIdx.x * 256 + threadIdx.x;
  if (i < n) p[i] = v;
}

// ---------------- degree accumulation (self-loop folded as deg init = 1.0) ----
__global__ __launch_bounds__(256) void k_degree(const int* __restrict__ dst,
                                                const float* __restrict__ w,
                                                float* __restrict__ deg, int e_) {
  int e = blockIdx.x * 256 + threadIdx.x;
  if (e < e_) atomicAdd(&deg[dst[e]], w[e]);
}

// ---------------- dinv (in place over deg), selfnorm, per-graph node count ----
__global__ __launch_bounds__(256) void k_dinv(float* __restrict__ deg,
                                              float* __restrict__ selfnorm,
                                              const int* __restrict__ batch,
                                              float* __restrict__ cnt, int n) {
  int i = blockIdx.x * 256 + threadIdx.x;
  if (i >= n) return;
  float d = deg[i];
  float di = (d > 0.f) ? __frsqrt_rn(d) : 0.f;
  deg[i] = di;                 // deg buffer now holds dinv
  selfnorm[i] = di * di;       // self-loop weight 1 * dinv^2
  atomicAdd(&cnt[batch[i]], 1.f);
}

// ---------------- per-edge normalization coefficient -------------------------
__global__ __launch_bounds__(256) void k_edgenorm(const int* __restrict__ src,
                                                  const int* __restrict__ dst,
                                                  const float* __restrict__ w,
                                                  const float* __restrict__ dinv,
                                                  float* __restrict__ norm, int e_) {
  int e = blockIdx.x * 256 + threadIdx.x;
  if (e < e_) norm[e] = dinv[src[e]] * w[e] * dinv[dst[e]];
}

// ---------------- WMMA fp32 GEMM: Y[N,64] = X[N,64] @ W[64,64] ----------------
// wave = 16 rows x 64 cols ; K split in 16 steps of 4 ; W staged in LDS
__global__ __launch_bounds__(256) void k_gemm_wmma(const float* __restrict__ X,
                                                   const float* __restrict__ W,
                                                   float* __restrict__ Y) {
  __shared__ float sW[HF * HF];
  for (int idx = threadIdx.x; idx < HF * HF; idx += 256) sW[idx] = W[idx];
  __syncthreads();

  const int wave = threadIdx.x >> 5;          // 0..7
  const int lane = threadIdx.x & 31;
  const int row0 = (blockIdx.x * 8 + wave) * 16;
  const int m    = lane & 15;
  const int kh   = (lane >> 4) << 1;          // 0 for lanes 0-15, 2 for 16-31

  v8f acc[4];
  #pragma unroll
  for (int t = 0; t < 4; ++t)
    #pragma unroll
    for (int r = 0; r < 8; ++r) acc[t][r] = 0.f;

  const float* xrow = X + (size_t)(row0 + m) * HF;
  #pragma unroll
  for (int k = 0; k < 16; ++k) {
    const int kk = 4 * k + kh;
    v2f a; a.x = xrow[kk]; a.y = xrow[kk + 1];        // A 16x4 fragment
    #pragma unroll
    for (int t = 0; t < 4; ++t) {
      v2f b;                                          // B 4x16 fragment
      b.x = sW[kk * HF + t * 16 + m];
      b.y = sW[(kk + 1) * HF + t * 16 + m];
      acc[t] = __builtin_amdgcn_wmma_f32_16x16x4_f32(
          false, a, false, b, (short)0, acc[t], false, false);
    }
  }

  const int rbase = row0 + ((lane >> 4) << 3);        // +8 for upper half-wave
  #pragma unroll
  for (int t = 0; t < 4; ++t)
    #pragma unroll
    for (int r = 0; r < 8; ++r)
      Y[(size_t)(rbase + r) * HF + t * 16 + m] = acc[t][r];
}

// ---------------- agg init: self-loop contribution + bias ---------------------
__global__ __launch_bounds__(256) void k_agg_init(const float* __restrict__ xw,
                                                  const float* __restrict__ selfnorm,
                                                  const float* __restrict__ bg,
                                                  float* __restrict__ agg) {
  int i = blockIdx.x * 256 + threadIdx.x;           // over N*64
  int n = i >> 6, f = i & 63;
  agg[i] = selfnorm[n] * xw[i] + bg[f];
}

// ---------------- edge scatter: 16 threads/edge, float4 per thread ------------
__global__ __launch_bounds__(256) void k_scatter(const float* __restrict__ xw,
                                                 float* __restrict__ agg,
                                                 const int* __restrict__ src,
                                                 const int* __restrict__ dst,
                                                 const float* __restrict__ norm, int e_) {
  int gid = blockIdx.x * 256 + threadIdx.x;
  int e = gid >> 4;
  if (e >= e_) return;
  int c = (gid & 15) << 2;
  float nv = norm[e];
  const float4 x = *(const float4*)(xw + (size_t)src[e] * HF + c);
  float* ad = agg + (size_t)dst[e] * HF + c;
  atomicAdd(ad + 0, nv * x.x);
  atomicAdd(ad + 1, nv * x.y);
  atomicAdd(ad + 2, nv * x.z);
  atomicAdd(ad + 3, nv * x.w);
}

// ---------------- per-graph sum / sumsq ---------------------------------------
__global__ __launch_bounds__(256) void k_stats(const float* __restrict__ agg,
                                               const int* __restrict__ batch,
                                               float* __restrict__ gsum,
                                               float* __restrict__ gsq) {
  int i = blockIdx.x * 256 + threadIdx.x;           // over N*64
  int n = i >> 6, f = i & 63;
  float v = agg[i];
  int g = batch[n];
  atomicAdd(&gsum[g * HF + f], v);
  atomicAdd(&gsq [g * HF + f], v * v);
}

// ---------------- finalize GraphNorm stats ------------------------------------
// var = E[x^2] - m^2*(2a - a^2);  store (a*m) and rsqrt(var+eps)
__global__ __launch_bounds__(256) void k_gn_final(const float* __restrict__ gsum,
                                                  const float* __restrict__ gsq,
                                                  const float* __restrict__ cnt,
                                                  const float* __restrict__ alpha,
                                                  float* __restrict__ gmean,
                                                  float* __restrict__ ginv) {
  int i = blockIdx.x * 256 + threadIdx.x;           // over G*64
  if (i >= N_GRAPH * HF) return;
  int g = i >> 6, f = i & 63;
  float c = cnt[g];
  float rc = (c > 0.f) ? 1.f / c : 0.f;
  float m = gsum[i] * rc;
  float q = gsq[i] * rc;
  float a = alpha[f];
  float var = q - m * m * (2.f * a - a * a);
  gmean[i] = a * m;
  ginv[i]  = __frsqrt_rn(var + EPSV);
}

// ---------------- normalize + relu + residual + max-pool ----------------------
__global__ __launch_bounds__(256) void k_norm_pool(const float* __restrict__ agg,
                                                   const int* __restrict__ batch,
                                                   const float* __restrict__ gmean,
                                                   const float* __restrict__ ginv,
                                                   const float* __restrict__ gamma,
                                                   const float* __restrict__ beta,
                                                   float* __restrict__ h,
                                                   float* __restrict__ pooled,
                                                   int add_res) {
  int i = blockIdx.x * 256 + threadIdx.x;           // over N*64
  int n = i >> 6, f = i & 63;
  int g = batch[n];
  float xc = agg[i] - gmean[g * HF + f];
  float y = gamma[f] * xc * ginv[g * HF + f] + beta[f];
  y = fmaxf(y, 0.f);
  if (add_res) y += h[i];
  h[i] = y;
  // y >= 0 always -> IEEE order == unsigned bit order; pooled pre-filled with +0.0
  atomicMax((unsigned int*)&pooled[(size_t)g * HF + f], __float_as_uint(y));
}

// ---------------- dense head --------------------------------------------------
__global__ __launch_bounds__(256) void k_head_in(const float* __restrict__ pooled, // [L][G][64]
                                                 const float* __restrict__ Wd,     // [256,64]
                                                 const float* __restrict__ bd,
                                                 float* __restrict__ z) {          // [G,64]
  int i = blockIdx.x * 256 + threadIdx.x;
  if (i >= N_GRAPH * HF) return;
  int g = i >> 6, j = i & 63;
  float acc = bd[j];
  #pragma unroll 4
  for (int k = 0; k < NLAY * HF; ++k) {
    float x = pooled[(size_t)(k >> 6) * N_GRAPH * HF + g * HF + (k & 63)];
    acc += x * Wd[k * HF + j];
  }
  z[i] = fmaxf(acc, 0.f);
}

__global__ __launch_bounds__(256) void k_head_mid(const float* __restrict__ zin,
                                                  const float* __restrict__ Wd, // [64,64]
                                                  const float* __restrict__ bd,
                                                  float* __restrict__ z) {
  int i = blockIdx.x * 256 + threadIdx.x;
  if (i >= N_GRAPH * HF) return;
  int g = i >> 6, j = i & 63;
  float acc = bd[j];
  #pragma unroll 8
  for (int k = 0; k < HF; ++k) acc += zin[g * HF + k] * Wd[k * HF + j];
  z[i] = fmaxf(acc, 0.f);
}

__global__ __launch_bounds__(256) void k_head_out(const float* __restrict__ zin,
                                                  const float* __restrict__ Wd, // [64,10]
                                                  const float* __restrict__ bd,
                                                  float* __restrict__ out) {
  int i = blockIdx.x * 256 + threadIdx.x;
  if (i >= N_GRAPH * NCLS) return;
  int g = i / NCLS, c = i % NCLS;
  float acc = bd[c];
  #pragma unroll 8
  for (int k = 0; k < HF; ++k) acc += zin[g * HF + k] * Wd[k * NCLS + c];
  out[i] = acc;
}

// =============================================================================
extern "C" void kernel_launch(void* const* d_in, const int* in_sizes, int n_in,
                              void* d_out, int out_size, void* d_ws, size_t ws_size,
                              hipStream_t stream) {
  const float* inputs = (const float*)d_in[0];
  const int*   ei     = (const int*)d_in[1];     // [2,E]: row0=src, row1=dst
  const int*   batch  = (const int*)d_in[2];
  const float* ew     = (const float*)d_in[3];
  const float* Wg     = (const float*)d_in[4];   // [L,64,64]
  const float* bg     = (const float*)d_in[5];   // [L,64]
  const float* gamma  = (const float*)d_in[6];
  const float* beta   = (const float*)d_in[7];
  const float* alpha  = (const float*)d_in[8];
  const float* Wd_in  = (const float*)d_in[9];   // [256,64]
  const float* bd_in  = (const float*)d_in[10];
  const float* Wd1    = (const float*)d_in[11];  // [64,64]
  const float* bd1    = (const float*)d_in[12];
  const float* Wd_out = (const float*)d_in[13];  // [64,10]
  const float* bd_out = (const float*)d_in[14];
  float* out = (float*)d_out;

  const int* src = ei;
  const int* dst = ei + N_EDGES;

  // ---- workspace carving (bytes, 1KB-aligned chunks) ----
  char* ws = (char*)d_ws;
  size_t off = 0;
  float* deg      = (float*)(ws + off); off += (size_t)N_NODES * 4;            // -> dinv
  float* cnt      = (float*)(ws + off); off += 1024;
  float* norm     = (float*)(ws + off); off += (size_t)N_EDGES * 4;
  float* selfnorm = (float*)(ws + off); off += (size_t)N_NODES * 4;
  float* hbuf     = (float*)(ws + off); off += (size_t)N_NODES * HF * 4;
  float* xw       = (float*)(ws + off); off += (size_t)N_NODES * HF * 4;
  float* agg      = (float*)(ws + off); off += (size_t)N_NODES * HF * 4;
  float* gsum     = (float*)(ws + off); off += (size_t)N_GRAPH * HF * 4;       // gsum,gsq adjacent
  float* gsq      = (float*)(ws + off); off += (size_t)N_GRAPH * HF * 4;
  float* gmean    = (float*)(ws + off); off += (size_t)N_GRAPH * HF * 4;
  float* ginv     = (float*)(ws + off); off += (size_t)N_GRAPH * HF * 4;
  float* pooled   = (float*)(ws + off); off += (size_t)NLAY * N_GRAPH * HF * 4;
  float* z1       = (float*)(ws + off); off += (size_t)N_GRAPH * HF * 4;
  float* z2       = (float*)(ws + off); off += (size_t)N_GRAPH * HF * 4;
  (void)ws_size; (void)n_in; (void)in_sizes; (void)out_size;

  const int NB_N   = (N_NODES + 255) / 256;
  const int NB_E   = (N_EDGES + 255) / 256;
  const int NB_NF  = (N_NODES * HF + 255) / 256;
  const int NB_GF  = (N_GRAPH * HF + 255) / 256;

  // ---- precompute (re-done every call: deterministic) ----
  k_fill<<<NB_N, 256, 0, stream>>>(deg, 1.0f, N_NODES);                 // self-loop w=1
  k_fill<<<1, 256, 0, stream>>>(cnt, 0.0f, N_GRAPH);
  k_fill<<<(NLAY * N_GRAPH * HF + 255) / 256, 256, 0, stream>>>(pooled, 0.0f,
                                                                NLAY * N_GRAPH * HF);
  k_degree<<<NB_E, 256, 0, stream>>>(dst, ew, deg, N_EDGES);
  k_dinv<<<NB_N, 256, 0, stream>>>(deg, selfnorm, batch, cnt, N_NODES); // deg := dinv
  k_edgenorm<<<NB_E, 256, 0, stream>>>(src, dst, ew, deg, norm, N_EDGES);

  // ---- GCN blocks ----
  for (int i = 0; i < NLAY; ++i) {
    const float* hin = (i == 0) ? inputs : hbuf;
    k_gemm_wmma<<<N_NODES / 128, 256, 0, stream>>>(hin, Wg + (size_t)i * HF * HF, xw);
    k_agg_init<<<NB_NF, 256, 0, stream>>>(xw, selfnorm, bg + i * HF, agg);
    k_scatter<<<(N_EDGES * 16) / 256, 256, 0, stream>>>(xw, agg, src, dst, norm, N_EDGES);
    k_fill<<<(2 * N_GRAPH * HF + 255) / 256, 256, 0, stream>>>(gsum, 0.0f, 2 * N_GRAPH * HF);
    k_stats<<<NB_NF, 256, 0, stream>>>(agg, batch, gsum, gsq);
    k_gn_final<<<NB_GF, 256, 0, stream>>>(gsum, gsq, cnt, alpha + i * HF, gmean, ginv);
    k_norm_pool<<<NB_NF, 256, 0, stream>>>(agg, batch, gmean, ginv, gamma + i * HF,
                                           beta + i * HF, hbuf,
                                           pooled + (size_t)i * N_GRAPH * HF,
                                           (i > 0) ? 1 : 0);
  }

  // ---- dense head ----
  k_head_in <<<NB_GF, 256, 0, stream>>>(pooled, Wd_in, bd_in, z1);
  k_head_mid<<<NB_GF, 256, 0, stream>>>(z1, Wd1, bd1, z2);
  k_head_out<<<(N_GRAPH * NCLS + 255) / 256, 256, 0, stream>>>(z2, Wd_out, bd_out, out);
}